// DataPreprocessor_60739427500336
// MI455X (gfx1250) — compile-verified
//
#include <hip/hip_runtime.h>
#include <stdint.h>

// Pure permutation kernel: out[b, z1*32+z2, p*16+q] = data[b, r&1, (r>>1)*512 + z2*16+q],
// r = z1*16+p.  All movement is 64-byte-block granular -> b128 async global<->LDS DMA.
//
// 2048 blocks x 256 threads, 16 float4 per thread = 33,554,432 floats = out_size.

#define THREADS 256
#define F4_PER_THREAD 16
#define TILE 8   // float4 per wait-batch per thread (2 batches, double-buffered LDS)

__global__ __launch_bounds__(THREADS) void DataPreprocessor_permute_async(
    const float* __restrict__ in, float* __restrict__ out)
{
    // 2 buffers * 8 slots * 256 threads * 16 B = 64 KB LDS (well under 320 KB/WGP)
    __shared__ alignas(16) unsigned char smem[2 * TILE * THREADS * 16];

    const unsigned t        = threadIdx.x;
    const unsigned ldsBase  = (unsigned)(uintptr_t)&smem[0];
    const uint64_t inBase   = (uint64_t)in;
    const uint64_t outBase  = (uint64_t)out;
    const unsigned blockV0  = blockIdx.x * (THREADS * F4_PER_THREAD); // first float4 idx

    #pragma unroll
    for (int tile = 0; tile < F4_PER_THREAD / TILE; ++tile) {
        const unsigned bufBase = ldsBase + (unsigned)(tile & 1) * (TILE * THREADS * 16);

        // Phase 1: gather 8 x 16B from input into LDS (output-ordered), async.
        #pragma unroll
        for (int k = 0; k < TILE; ++k) {
            const unsigned v     = blockV0 + (unsigned)(tile * TILE + k) * THREADS + t;
            const unsigned q4    = v & 3u;        // float4 within the 16-float chunk
            const unsigned chunk = v >> 2;        // 16-float (64B) chunk index
            const unsigned p  = chunk & 15u;
            const unsigned z2 = (chunk >> 4) & 31u;
            const unsigned z1 = (chunk >> 9) & 15u;
            const unsigned b  = chunk >> 13;
            const unsigned r  = (z1 << 4) + p;    // interleaved row 0..255
            // src float4 index = b*32768 + (r&1)*16384 + (r>>1)*128 + z2*4 + q4
            const unsigned src4   = (((b << 1) + (r & 1u)) << 14) + ((r >> 1) << 7)
                                    + (z2 << 2) + q4;
            const unsigned srcOff = src4 << 4;                                   // bytes
            const unsigned ldsOff = bufBase + ((unsigned)(k * THREADS) + t) * 16u;
            asm volatile("global_load_async_to_lds_b128 %0, %1, %2"
                         :
                         : "v"(ldsOff), "v"(srcOff), "s"(inBase)
                         : "memory");
        }

        // Drain this wave's outstanding async ops (loads above + prior tile stores).
        asm volatile("s_wait_asynccnt 0" ::: "memory");

        // Phase 2: stream LDS out to the contiguous destination, async.
        #pragma unroll
        for (int k = 0; k < TILE; ++k) {
            const unsigned v      = blockV0 + (unsigned)(tile * TILE + k) * THREADS + t;
            const unsigned dstOff = v << 4;                                      // bytes
            const unsigned ldsOff = bufBase + ((unsigned)(k * THREADS) + t) * 16u;
            asm volatile("global_store_async_from_lds_b128 %0, %1, %2"
                         :
                         : "v"(dstOff), "v"(ldsOff), "s"(outBase)
                         : "memory");
        }
    }

    // Make sure all async stores retired before wave teardown.
    asm volatile("s_wait_asynccnt 0" ::: "memory");
}

extern "C" void kernel_launch(void* const* d_in, const int* in_sizes, int n_in,
                              void* d_out, int out_size, void* d_ws, size_t ws_size,
                              hipStream_t stream) {
    (void)in_sizes; (void)n_in; (void)d_ws; (void)ws_size;
    const float* in  = (const float*)d_in[0];
    float*       out = (float*)d_out;

    const unsigned totalF4 = (unsigned)(out_size / 4);              // 8,388,608
    const unsigned blocks  = totalF4 / (THREADS * F4_PER_THREAD);   // 2048

    DataPreprocessor_permute_async<<<blocks, THREADS, 0, stream>>>(in, out);
}